// SBGMN_60120952209417
// MI455X (gfx1250) — compile-verified
//
#include <hip/hip_runtime.h>
#include <hip/hip_bf16.h>

#define DIM 128

typedef float v2f __attribute__((ext_vector_type(2)));
typedef float v8f __attribute__((ext_vector_type(8)));

// ---------------------------------------------------------------------------
// Kernel 1: new_emb = feature_b @ W_mlp^T + b_mlp    (f32 WMMA 16x16x4)
// C[i,j] = sum_k feature_b[i,k] * W_mlp[j,k]
// A-frag (16x4 f32, 2 VGPRs):  lane l: M = l&15, K-half = (l>=16)?2:0,
//                              v0/v1 = consecutive K -> one b64 load.
// B-frag (4x16 f32, 2 VGPRs):  lane l: N = l&15, same K-half scheme; since
//                              B[k][n] = W_mlp[n][k], consecutive K is a
//                              contiguous b64 load from W_mlp's row n.
// C/D (16x16 f32, 8 VGPRs):    vgpr r: M = r + ((l>=16)?8:0), N = l&15.
// ---------------------------------------------------------------------------
__global__ void wmma_gemm_bias(const float* __restrict__ featB,
                               const float* __restrict__ W,
                               const float* __restrict__ bias,
                               float* __restrict__ newEmb,
                               int nb) {
    const int wavesPerBlock = blockDim.x >> 5;
    const int waveId = threadIdx.x >> 5;
    const int lane   = threadIdx.x & 31;
    const int tile   = blockIdx.x * wavesPerBlock + waveId;

    const int nTiles = DIM / 16;            // 8 tiles along N
    const int mTile  = tile / nTiles;
    const int nTile  = tile % nTiles;
    if (mTile * 16 >= nb) return;

    const int mn    = lane & 15;            // M for A-frag, N for B-frag
    const int khalf = (lane >> 4) << 1;     // 0 or 2

    const float* aRow = featB + (size_t)(mTile * 16 + mn) * DIM + khalf;
    const float* bRow = W     + (size_t)(nTile * 16 + mn) * DIM + khalf;

    v8f acc = {};
#pragma unroll
    for (int k = 0; k < DIM; k += 4) {
        v2f a = *(const v2f*)(aRow + k);
        v2f b = *(const v2f*)(bRow + k);
        acc = __builtin_amdgcn_wmma_f32_16x16x4_f32(
            /*neg_a=*/false, a, /*neg_b=*/false, b,
            /*c_mod=*/(short)0, acc, /*reuse_a=*/false, /*reuse_b=*/false);
    }

    const float bv = bias[nTile * 16 + mn];
    float* outCol = newEmb + (size_t)(mTile * 16) * DIM + nTile * 16 + mn;
    const int mOff = (lane >> 4) << 3;      // 0 or 8
#pragma unroll
    for (int r = 0; r < 8; ++r) {
        outCol[(size_t)(r + mOff) * DIM] = acc[r] + bv;
    }
}

// ---------------------------------------------------------------------------
// Kernel 2: out[i] = dot(X[i,:], v)   one wave per row, float4 per lane
// ---------------------------------------------------------------------------
__global__ void row_dot(const float* __restrict__ X,
                        const float* __restrict__ v,
                        float* __restrict__ out, int n) {
    const int wave = blockIdx.x * (blockDim.x >> 5) + (threadIdx.x >> 5);
    const int lane = threadIdx.x & 31;
    if (wave >= n) return;
    const float4 x = ((const float4*)(X + (size_t)wave * DIM))[lane];
    const float4 a = ((const float4*)v)[lane];
    float s = x.x * a.x + x.y * a.y + x.z * a.z + x.w * a.w;
#pragma unroll
    for (int off = 16; off > 0; off >>= 1)
        s += __shfl_down(s, off, 32);
    if (lane == 0) out[wave] = s;
}

// ---------------------------------------------------------------------------
// Kernel 3: per-edge attention weight + scatter-add.
// One wave per edge; lane handles a float4 slice of the 128-wide message.
// agg (= d_out) and row_sum must be pre-zeroed.
// ---------------------------------------------------------------------------
__global__ void edge_scatter(const long long* __restrict__ edges,
                             const float* __restrict__ sa,
                             const float* __restrict__ sb,
                             const float* __restrict__ newEmb,
                             float* __restrict__ agg,
                             float* __restrict__ rowSum,
                             int ne) {
    const int wave = blockIdx.x * (blockDim.x >> 5) + (threadIdx.x >> 5);
    const int lane = threadIdx.x & 31;
    if (wave >= ne) return;

    const long long src = edges[2LL * wave];
    const long long dst = edges[2LL * wave + 1];

    const float score = sa[src] + sb[dst];
    const float elu   = (score > 0.0f) ? score : 0.1f * expm1f(score);
    const float w     = expf(elu);

    if (lane == 0) atomicAdd(rowSum + src, w);

    const float4 e = ((const float4*)(newEmb + (size_t)dst * DIM))[lane];
    float* o = agg + (size_t)src * DIM + lane * 4;
    atomicAdd(o + 0, w * e.x);
    atomicAdd(o + 1, w * e.y);
    atomicAdd(o + 2, w * e.z);
    atomicAdd(o + 3, w * e.w);
}

// ---------------------------------------------------------------------------
// Kernel 4: out = agg / (row_sum == 0 ? 1 : row_sum)
// ---------------------------------------------------------------------------
__global__ void finalize_div(float* __restrict__ agg,
                             const float* __restrict__ rowSum, int total) {
    const int i = blockIdx.x * blockDim.x + threadIdx.x;
    if (i >= total) return;
    const float rs = rowSum[i >> 7];   // DIM == 128
    agg[i] = agg[i] / ((rs == 0.0f) ? 1.0f : rs);
}

// ---------------------------------------------------------------------------
extern "C" void kernel_launch(void* const* d_in, const int* in_sizes, int n_in,
                              void* d_out, int out_size, void* d_ws, size_t ws_size,
                              hipStream_t stream) {
    const float*     feature_a = (const float*)d_in[0];
    const float*     feature_b = (const float*)d_in[1];
    const long long* edges     = (const long long*)d_in[2];
    const float*     W_mlp     = (const float*)d_in[3];
    const float*     b_mlp     = (const float*)d_in[4];
    const float*     a_vec     = (const float*)d_in[5];

    const int NA = in_sizes[0] / DIM;
    const int NB = in_sizes[1] / DIM;
    const int E  = in_sizes[2] / 2;

    float* agg = (float*)d_out;                    // NA x DIM

    // workspace carve-up
    float* new_emb = (float*)d_ws;                 // NB x DIM
    float* s_a     = new_emb + (size_t)NB * DIM;   // NA
    float* s_b     = s_a + NA;                     // NB
    float* row_sum = s_b + NB;                     // NA

    const float* a_top = a_vec;                    // a_vec[:DIM]
    const float* a_bot = a_vec + DIM;              // a_vec[DIM:]

    // zero accumulators (graph-capture-safe memsets)
    hipMemsetAsync(agg,     0, (size_t)NA * DIM * sizeof(float), stream);
    hipMemsetAsync(row_sum, 0, (size_t)NA * sizeof(float),       stream);

    // 1) GEMM: new_emb = feature_b @ W^T + b   (one wave per 16x16 tile)
    {
        const int tiles  = ((NB + 15) / 16) * (DIM / 16);
        const int wavesPerBlock = 8;               // 256 threads
        const int blocks = (tiles + wavesPerBlock - 1) / wavesPerBlock;
        wmma_gemm_bias<<<blocks, 256, 0, stream>>>(feature_b, W_mlp, b_mlp,
                                                   new_emb, NB);
    }

    // 2) s_a = feature_a @ a_top ; s_b = new_emb @ a_bot
    {
        const int wpb = 8;
        row_dot<<<(NA + wpb - 1) / wpb, 256, 0, stream>>>(feature_a, a_top, s_a, NA);
        row_dot<<<(NB + wpb - 1) / wpb, 256, 0, stream>>>(new_emb,  a_bot, s_b, NB);
    }

    // 3) per-edge attention + scatter
    {
        const int wpb = 8;
        edge_scatter<<<(E + wpb - 1) / wpb, 256, 0, stream>>>(
            edges, s_a, s_b, new_emb, agg, row_sum, E);
    }

    // 4) normalize
    {
        const int total = NA * DIM;
        finalize_div<<<(total + 255) / 256, 256, 0, stream>>>(agg, row_sum, total);
    }
}